// AR_19782619365450
// MI455X (gfx1250) — compile-verified
//
#include <hip/hip_runtime.h>
#include <stdint.h>

// AR(1) scan  x_t = 0.8*x_{t-1} + 0.1*eps_t  on CDNA5 (gfx1250).
//
// Per 16-row x 16-step tile:  D[m,n] = c^(n+1)*carry[m] + sum_j EPS[m,j]*M[j,n]
// with M[j,n] = STD*c^(n-j) (n>=j)  ->  4 x V_WMMA_F32_16X16X4_F32, carry
// folded into the accumulator init, carry-out = D[:,15] via ds_bpermute.
//
// eps tiles staged global->LDS with GLOBAL_LOAD_ASYNC_TO_LDS_B128 (double-
// buffered, ASYNCcnt-synchronized, in-order completion), read back in WMMA
// A-layout via conflict-free ds_load_b64 (LDS rows padded to 20 floats).
//
// Time axis parallelized by exponential forgetting (0.8^256 ~ 1.6e-25 << f32
// ulp): 4 chunks, each warming up 256 steps from zero state (chunk 0 starts
// from the true x0). 512 row-blocks x 4 chunks = 2048 waves.

typedef float v2f __attribute__((ext_vector_type(2)));
typedef float v8f __attribute__((ext_vector_type(8)));

namespace {
constexpr float kCoeff = 0.8f;
constexpr float kStd   = 0.1f;
constexpr int kB = 8192;
constexpr int kT = 4096;                 // output columns
constexpr int kE = kT - 1;               // 4095 eps columns
constexpr int kTiles = 256;              // tile 255 has 15 valid cols
constexpr int kChunks = 4;
constexpr int kTPC = kTiles / kChunks;   // 64 tiles per chunk
constexpr int kWarm = 16;                // 256 warm-up steps
constexpr int kWavesPerBlock = 8;
constexpr int kLdsRowF = 20;             // padded LDS row stride (floats)
constexpr int kBufF = 16 * kLdsRowF;     // 320 floats per tile buffer
}

__device__ __forceinline__ float cpow(int e) {  // kCoeff^e, 0 <= e <= 16
    float p = 1.0f;
#pragma unroll
    for (int i = 0; i < 16; ++i) p = (i < e) ? p * kCoeff : p;
    return p;
}

__global__ __launch_bounds__(kWavesPerBlock * 32) void ar1_wmma(
        const float* __restrict__ x0,
        const float* __restrict__ eps,
        float* __restrict__ out) {
    __shared__ float smem[kWavesPerBlock][2][kBufF];   // 20 KB / block

    const int lane  = threadIdx.x & 31;
    const int wslot = threadIdx.x >> 5;
    // Scalarize all wave-uniform control state: bounds -> SGPRs, guards ->
    // s_cmp/s_cbranch instead of v_cmpx + exec juggling.
    const int wave  = __builtin_amdgcn_readfirstlane(
                          blockIdx.x * kWavesPerBlock + wslot);
    const int nRB   = kB / 16;           // 512 row blocks
    const int rb    = wave & (nRB - 1);
    const int chunk = wave / nRB;        // 0..3 exact

    const int m    = lane & 15;          // A row / D column index n
    const int hi   = lane >> 4;          // half-wave: K pair (A/B), M+8 (C/D)
    const int row0 = rb * 16;

    // Constant triangular B slices: B_s[k,n] = STD*c^(n-k), n>=k.
    v2f bmat[4];
#pragma unroll
    for (int s = 0; s < 4; ++s) {
        const int k0 = 4 * s + 2 * hi;
        bmat[s].x = (m >= k0)     ? kStd * cpow(m - k0)     : 0.0f;
        bmat[s].y = (m >= k0 + 1) ? kStd * cpow(m - k0 - 1) : 0.0f;
    }
    const float pv = cpow(m + 1);        // carry weights c^(n+1)

    // Carry in C/D broadcast layout: carry_b[r] = carry[row r + 8*hi].
    float carry_b[8];
#pragma unroll
    for (int r = 0; r < 8; ++r)
        carry_b[r] = (chunk == 0) ? x0[row0 + r + 8 * hi] : 0.0f;

    if (chunk == 0 && hi == 0)           // output column 0 is x0
        out[(size_t)(row0 + m) * kT] = x0[row0 + m];

    // ---- async global->LDS staging addresses ----
    // Lane pair (lr, lq): 16B of row lr, float4 group lq; second instruction
    // (+32B both sides) covers groups 2..3 of the same row.
    const int lr = lane >> 1;
    const int lq = lane & 1;

    const int gFirst = chunk * kTPC - (chunk ? kWarm : 0);
    const int gStore = chunk * kTPC;
    // chunk 3's last (partial) tile is peeled off the pipelined loop.
    const int gMain  = (chunk + 1) * kTPC - (chunk == kChunks - 1 ? 1 : 0);

    // Double-buffer pointers, swapped per tile (renames after unrolling).
    const float* rdCur = &smem[wslot][0][m * kLdsRowF];
    const float* rdNxt = &smem[wslot][1][m * kLdsRowF];
    uint32_t wrCur =
        (uint32_t)(uintptr_t)(&smem[wslot][0][lr * kLdsRowF + 4 * lq]);
    uint32_t wrNxt = wrCur + (uint32_t)(kBufF * sizeof(float));

    // Incrementing global addresses: +64 B (16 floats) per tile.
    uint64_t gaPf = (uint64_t)(uintptr_t)(eps + (size_t)(row0 + lr) * kE + 4 * lq)
                  + (uint64_t)(gFirst * 16) * 4;
    float* op = out + (size_t)(row0 + 8 * hi) * kT + (size_t)(gStore * 16) + 1 + m;

    // Prologue: prefetch tiles gFirst (buf0) and gFirst+1 (buf1).
    asm volatile(
        "global_load_async_to_lds_b128 %0, %1, off\n\t"
        "global_load_async_to_lds_b128 %0, %1, off offset:32"
        :: "v"(wrCur), "v"(gaPf) : "memory");
    asm volatile(
        "global_load_async_to_lds_b128 %0, %1, off\n\t"
        "global_load_async_to_lds_b128 %0, %1, off offset:32"
        :: "v"(wrNxt), "v"(gaPf + 64) : "memory");
    gaPf += 128;                         // next tile to prefetch: gFirst+2

    const int bsrc = 15 | (lane & 16);   // carry broadcast source lane

    auto tile = [&](int g, bool doStore, bool guarded) __attribute__((always_inline)) {
        // In-order async completion: allow the next tile's 2 loads to stay
        // outstanding except when nothing is behind ours.
        if (!guarded || g + 1 < gMain)
            asm volatile("s_wait_asynccnt 0x2" ::: "memory");
        else
            asm volatile("s_wait_asynccnt 0x0" ::: "memory");

        v2f a[4];
#pragma unroll
        for (int s = 0; s < 4; ++s)
            a[s] = *(const v2f*)(rdCur + 4 * s + 2 * hi);

        v8f acc;
#pragma unroll
        for (int r = 0; r < 8; ++r) acc[r] = carry_b[r] * pv;
#pragma unroll
        for (int s = 0; s < 4; ++s)
            acc = __builtin_amdgcn_wmma_f32_16x16x4_f32(
                false, a[s], false, bmat[s], (short)0, acc, false, false);

        // Prefetch tile g+2 into the buffer we just finished reading.
        if (!guarded || g + 2 < gMain) {
            asm volatile(
                "s_wait_dscnt 0x0\n\t"
                "global_load_async_to_lds_b128 %0, %1, off\n\t"
                "global_load_async_to_lds_b128 %0, %1, off offset:32"
                :: "v"(wrCur), "v"(gaPf) : "memory");
        }
        gaPf += 64;

        // New carry = D[:,15]: lanes 0-15 <- lane 15, lanes 16-31 <- lane 31.
#pragma unroll
        for (int r = 0; r < 8; ++r) carry_b[r] = __shfl(acc[r], bsrc, 32);

        if (doStore) {
#pragma unroll
            for (int r = 0; r < 8; ++r) op[(size_t)r * kT] = acc[r];
            op += 16;
        }

        const float* tr = rdCur; rdCur = rdNxt; rdNxt = tr;
        const uint32_t tw = wrCur; wrCur = wrNxt; wrNxt = tw;
    };

    // Warm-up (chunks 1-3): evolve carry only; prefetch always in range.
    for (int g = gFirst; g < gStore; ++g)
        tile(g, false, false);
    // Main store loop.
    for (int g = gStore; g < gMain; ++g)
        tile(g, true, true);

    // Peeled tail tile (chunk 3 only): eps col 4095 / out col 4096 don't
    // exist. Direct guarded loads, masked store.
    if (chunk == kChunks - 1) {
        const int ec = (kTiles - 1) * 16;              // 4080
        const float* __restrict__ epsRow = eps + (size_t)(row0 + m) * kE;
        v2f a[4];
#pragma unroll
        for (int s = 0; s < 4; ++s) {
            const int c0 = ec + 4 * s + 2 * hi;        // max 4094, in range
            a[s].x = epsRow[c0];
            a[s].y = (c0 + 1 < kE) ? epsRow[c0 + 1] : 0.0f;
        }
        v8f acc;
#pragma unroll
        for (int r = 0; r < 8; ++r) acc[r] = carry_b[r] * pv;
#pragma unroll
        for (int s = 0; s < 4; ++s)
            acc = __builtin_amdgcn_wmma_f32_16x16x4_f32(
                false, a[s], false, bmat[s], (short)0, acc, false, false);
        const int oc = ec + 1 + m;                     // 4081 + m
        if (oc < kT) {
#pragma unroll
            for (int r = 0; r < 8; ++r)
                out[(size_t)(row0 + r + 8 * hi) * kT + oc] = acc[r];
        }
    }
}

extern "C" void kernel_launch(void* const* d_in, const int* in_sizes, int n_in,
                              void* d_out, int out_size, void* d_ws, size_t ws_size,
                              hipStream_t stream) {
    (void)in_sizes; (void)n_in; (void)out_size; (void)d_ws; (void)ws_size;
    const float* x0  = (const float*)d_in[0];   // [B, 1]
    const float* eps = (const float*)d_in[1];   // [B, T-1]
    float* out = (float*)d_out;                 // [B, T]

    // 512 row-blocks * 4 time-chunks = 2048 waves = 65536 threads.
    const int threads = kWavesPerBlock * 32;                 // 256
    const int blocks  = (kB / 16) * kChunks * 32 / threads;  // 256
    ar1_wmma<<<blocks, threads, 0, stream>>>(x0, eps, out);
}